// BatchedMHA_71210557768221
// MI455X (gfx1250) — compile-verified
//
#include <hip/hip_runtime.h>
#include <hip/hip_bf16.h>
#include <math.h>

// ---------------------------------------------------------------------------
// Problem constants (from the reference)
// ---------------------------------------------------------------------------
#define HID_  2048
#define H_    16
#define DH_   128
#define L_    4096

typedef __attribute__((ext_vector_type(2))) float v2f;
typedef __attribute__((ext_vector_type(8))) float v8f;

// ===========================================================================
// GEMM: C[M,N] = A[M,K] @ W[N,K]^T   (row-major, K mult of 16, N mult of 128).
// Block = 256 threads = 8 waves (4x2), block tile 128x128, wave tile 32x64
// (2x4 of 16x16 V_WMMA_F32_16X16X4_F32 tiles).  Fragments are loaded from LDS
// as single 8-byte v2f loads (the f32 WMMA A/B layout keeps each lane's two
// elements contiguous), so the backend can emit ds_load_b64 into aligned
// VGPR pairs without repacking movs.
// ===========================================================================
#define GEMM_BM 128
#define GEMM_BN 128
#define GEMM_KC 16
#define LDX 20
#define LDW 20

__global__ __launch_bounds__(256)
void gemm_xwt_kernel(const float* __restrict__ A, const float* __restrict__ W,
                     float* __restrict__ C, int M, int N, int K) {
    __shared__ float Xs[GEMM_BM * LDX];
    __shared__ float Ws[GEMM_BN * LDW];

    const int tid  = threadIdx.x;
    const int lane = tid & 31;
    const int wave = tid >> 5;
    const int hi   = lane >> 4;     // which 16-lane half
    const int lm   = lane & 15;
    const int wm   = wave >> 1;     // 0..3 (M direction)
    const int wn   = wave & 1;      // 0..1 (N direction)

    const int blockM = blockIdx.y * GEMM_BM;
    const int blockN = blockIdx.x * GEMM_BN;
    const int m0 = blockM + wm * 32;
    const int n0 = blockN + wn * 64;

    // staging indices: 512 float4 per tile, 2 per thread
    const int row0 = tid >> 1;              // 0..127
    const int c40  = (tid & 1) * 2;         // float4 column 0 or 2

    v8f acc[2][4] = {};

    for (int kb = 0; kb < K; kb += GEMM_KC) {
        // ---- stage X tile (128 x 16) ----
        {
            int gm = blockM + row0;
            gm = gm < M ? gm : M - 1;                     // clamp, no divergence
            const float* src = A + (size_t)gm * K + kb + c40 * 4;
            *(float4*)(Xs + row0 * LDX + c40 * 4)       = *(const float4*)(src);
            *(float4*)(Xs + row0 * LDX + (c40 + 1) * 4) = *(const float4*)(src + 4);
            if (kb + GEMM_KC < K) __builtin_prefetch(src + GEMM_KC, 0, 1);
        }
        // ---- stage W tile (128 x 16) ----
        {
            int gn = blockN + row0;
            const float* src = W + (size_t)gn * K + kb + c40 * 4;
            *(float4*)(Ws + row0 * LDW + c40 * 4)       = *(const float4*)(src);
            *(float4*)(Ws + row0 * LDW + (c40 + 1) * 4) = *(const float4*)(src + 4);
            if (kb + GEMM_KC < K) __builtin_prefetch(src + GEMM_KC, 0, 1);
        }
        __syncthreads();

        #pragma unroll
        for (int ks = 0; ks < GEMM_KC; ks += 4) {
            v2f afrag[2], bfrag[4];
            #pragma unroll
            for (int i = 0; i < 2; ++i) {
                const int m = wm * 32 + i * 16 + lm;      // local row
                afrag[i] = *(const v2f*)(Xs + m * LDX + ks + 2 * hi);
            }
            #pragma unroll
            for (int j = 0; j < 4; ++j) {
                const int n = wn * 64 + j * 16 + lm;
                bfrag[j] = *(const v2f*)(Ws + n * LDW + ks + 2 * hi);
            }
            #pragma unroll
            for (int i = 0; i < 2; ++i)
                #pragma unroll
                for (int j = 0; j < 4; ++j)
                    acc[i][j] = __builtin_amdgcn_wmma_f32_16x16x4_f32(
                        false, afrag[i], false, bfrag[j],
                        (short)0, acc[i][j], false, false);
        }
        __syncthreads();
    }

    // ---- store (guarded; all WMMA already issued) ----
    #pragma unroll
    for (int i = 0; i < 2; ++i)
        #pragma unroll
        for (int j = 0; j < 4; ++j)
            #pragma unroll
            for (int v = 0; v < 8; ++v) {
                const int m = m0 + i * 16 + v + (hi ? 8 : 0);
                const int n = n0 + j * 16 + lm;
                if (m < M) C[(size_t)m * N + n] = acc[i][j][v];
            }
}

// ===========================================================================
// Decode attention: one block per (decode token, head), 128 threads.
// Flash-decode with online softmax; the freshly projected K/V row replaces
// cache position `seq_len` (the reference's cache .set) without mutating the
// input caches.  Pure bandwidth: coalesced float4 streaming of K/V.
// ===========================================================================
__global__ __launch_bounds__(128)
void decode_attn_kernel(const float* __restrict__ Qb, const float* __restrict__ Kb,
                        const float* __restrict__ Vb,
                        const float* __restrict__ kcache, const float* __restrict__ vcache,
                        const int* __restrict__ dlens, const int* __restrict__ didx,
                        float* __restrict__ O, float scale) {
    __shared__ float qs[DH_];
    __shared__ float sbuf[128];
    __shared__ float red[4];

    const int i = blockIdx.x / H_;   // decode token
    const int h = blockIdx.x % H_;
    const int t = threadIdx.x;       // owns output dim t

    const int col = dlens[i];
    const int b   = didx[i];
    const int P   = col + 1;

    qs[t] = Qb[(size_t)i * HID_ + h * DH_ + t];
    __syncthreads();

    const float* kbase = kcache + (((size_t)b * H_ + h) * L_) * DH_;
    const float* vbase = vcache + (((size_t)b * H_ + h) * L_) * DH_;
    const float* knew  = Kb + (size_t)i * HID_ + h * DH_;
    const float* vnew  = Vb + (size_t)i * HID_ + h * DH_;

    float m_run = -1e30f, l_run = 0.0f, acc = 0.0f;

    for (int c0 = 0; c0 < P; c0 += 128) {
        const int pos = c0 + t;
        float sv = -1e30f;
        if (pos < P) {
            const float* kr = (pos == col) ? knew : (kbase + (size_t)pos * DH_);
            float d = 0.0f;
            #pragma unroll
            for (int k = 0; k < DH_; k += 4) {
                float4 a  = *(const float4*)(qs + k);
                float4 kv = *(const float4*)(kr + k);
                d += a.x * kv.x + a.y * kv.y + a.z * kv.z + a.w * kv.w;
            }
            sv = d * scale;
        }
        // block max
        float bm = sv;
        #pragma unroll
        for (int msk = 16; msk; msk >>= 1) bm = fmaxf(bm, __shfl_xor(bm, msk));
        if ((t & 31) == 0) red[t >> 5] = bm;
        __syncthreads();
        bm = fmaxf(fmaxf(red[0], red[1]), fmaxf(red[2], red[3]));
        const float nm = fmaxf(m_run, bm);

        const float p = (pos < P) ? __expf(sv - nm) : 0.0f;
        sbuf[t] = p;
        float ps = p;
        #pragma unroll
        for (int msk = 16; msk; msk >>= 1) ps += __shfl_xor(ps, msk);
        __syncthreads();                 // everyone done reading red (max)
        if ((t & 31) == 0) red[t >> 5] = ps;
        __syncthreads();                 // sbuf + red (sum) visible
        ps = red[0] + red[1] + red[2] + red[3];

        const float f = __expf(m_run - nm);
        acc   *= f;
        l_run  = l_run * f + ps;
        m_run  = nm;

        const int cn = (P - c0 < 128) ? (P - c0) : 128;
        for (int j = 0; j < cn; ++j) {
            const int pj = c0 + j;
            const float* vr = (pj == col) ? vnew : (vbase + (size_t)pj * DH_);
            acc += sbuf[j] * vr[t];      // coalesced across the 128 threads
        }
        __syncthreads();                 // before sbuf/red reuse
    }

    O[(size_t)i * HID_ + h * DH_ + t] = acc / l_run;
}

// ===========================================================================
// Prefill attention: flash attention with WMMA.  One wave (32 threads) per
// (global 16-query tile, head).  Causal within each prefill sequence.
// ===========================================================================
#define QLD 132   // padded row stride (floats) for 16x128 tiles (even, 16B ok)
#define SLD 18    // P-tile stride (even -> v2f loads stay 8B aligned)

__global__ __launch_bounds__(32)
void prefill_attn_kernel(const float* __restrict__ Qb, const float* __restrict__ Kb,
                         const float* __restrict__ Vb,
                         const int* __restrict__ plens, int np, int nd,
                         float* __restrict__ O, float scale) {
    __shared__ float Qs[16 * QLD];
    __shared__ float Ks[16 * QLD];
    __shared__ float Vs[16 * QLD];
    __shared__ float Ss[16 * SLD];

    const int gqt = blockIdx.x;          // global query tile (16 rows)
    const int h   = blockIdx.y;

    // map global tile -> (sequence, local offset); tiles never straddle seqs
    int start = 0;
    {
        const int q0 = gqt * 16;
        int accum = 0;
        for (int s = 0; s < np; ++s) {
            const int len = plens[s];
            if (q0 < accum + len) { start = accum; break; }
            accum += len;
        }
    }
    const int t0 = nd + start;           // first global token of this sequence
    const int lq = gqt * 16 - start;     // local query row base within seq
    const int qt = lq >> 4;              // local query tile index

    const int lane = threadIdx.x;
    const int hi   = lane >> 4;
    const int lm   = lane & 15;

    // ---- load Q tile (16 x 128) into LDS ----
    for (int i = 0; i < 16; ++i) {
        float4 v = *(const float4*)(Qb + (size_t)(t0 + lq + i) * HID_ + h * DH_ + lane * 4);
        *(float4*)(Qs + i * QLD + lane * 4) = v;
    }

    v8f o[8] = {};
    float m_run[8], l_run[8];
    #pragma unroll
    for (int v = 0; v < 8; ++v) { m_run[v] = -1e30f; l_run[v] = 0.0f; }

    for (int kt = 0; kt <= qt; ++kt) {
        __syncthreads();
        // ---- stage K and V tiles (16 x 128 each) ----
        for (int i = 0; i < 16; ++i) {
            const size_t row = (size_t)(t0 + kt * 16 + i) * HID_ + h * DH_;
            *(float4*)(Ks + i * QLD + lane * 4) = *(const float4*)(Kb + row + lane * 4);
            *(float4*)(Vs + i * QLD + lane * 4) = *(const float4*)(Vb + row + lane * 4);
        }
        __syncthreads();

        // ---- S = Q * K^T  (16x16, accumulate over DH in K=4 steps) ----
        v8f s = {};
        #pragma unroll
        for (int ks = 0; ks < DH_; ks += 4) {
            v2f a   = *(const v2f*)(Qs + lm * QLD + ks + 2 * hi);
            v2f bfr = *(const v2f*)(Ks + lm * QLD + ks + 2 * hi);  // B[k][n]=K[n][k]
            s = __builtin_amdgcn_wmma_f32_16x16x4_f32(false, a, false, bfr,
                                                      (short)0, s, false, false);
        }

        // ---- scale + causal mask (diagonal tile only) ----
        #pragma unroll
        for (int v = 0; v < 8; ++v) {
            float sv = s[v] * scale;
            const int m = v + (hi ? 8 : 0);
            if (kt == qt && lm > m) sv = -1e30f;
            s[v] = sv;
        }

        // ---- online softmax, per row (rows live on 16-lane halves) ----
        float fsc[8];
        #pragma unroll
        for (int v = 0; v < 8; ++v) {
            float rm = s[v];
            #pragma unroll
            for (int msk = 8; msk; msk >>= 1) rm = fmaxf(rm, __shfl_xor(rm, msk));
            const float nm = fmaxf(m_run[v], rm);
            const float f  = __expf(m_run[v] - nm);
            const float p  = __expf(s[v] - nm);
            float rs = p;
            #pragma unroll
            for (int msk = 8; msk; msk >>= 1) rs += __shfl_xor(rs, msk);
            l_run[v] = l_run[v] * f + rs;
            m_run[v] = nm;
            s[v]     = p;
            fsc[v]   = f;
        }
        #pragma unroll
        for (int nb = 0; nb < 8; ++nb)
            #pragma unroll
            for (int v = 0; v < 8; ++v) o[nb][v] *= fsc[v];

        // ---- relayout P through LDS (C layout -> A layout) ----
        #pragma unroll
        for (int v = 0; v < 8; ++v)
            Ss[(v + (hi ? 8 : 0)) * SLD + lm] = s[v];
        __syncthreads();

        // ---- O += P * V  (8 N-subtiles of 16 dims, 4 k-steps) ----
        #pragma unroll
        for (int ks = 0; ks < 16; ks += 4) {
            v2f a = *(const v2f*)(Ss + lm * SLD + ks + 2 * hi);
            #pragma unroll
            for (int nb = 0; nb < 8; ++nb) {
                v2f bfr;
                bfr[0] = Vs[(ks + 2 * hi)     * QLD + nb * 16 + lm];
                bfr[1] = Vs[(ks + 2 * hi + 1) * QLD + nb * 16 + lm];
                o[nb] = __builtin_amdgcn_wmma_f32_16x16x4_f32(false, a, false, bfr,
                                                              (short)0, o[nb], false, false);
            }
        }
    }

    // ---- epilogue: normalize and store ----
    #pragma unroll
    for (int nb = 0; nb < 8; ++nb)
        #pragma unroll
        for (int v = 0; v < 8; ++v) {
            const int m = v + (hi ? 8 : 0);
            O[(size_t)(t0 + lq + m) * HID_ + h * DH_ + nb * 16 + lm] = o[nb][v] / l_run[v];
        }
}

// ===========================================================================
// Launch
// ===========================================================================
extern "C" void kernel_launch(void* const* d_in, const int* in_sizes, int n_in,
                              void* d_out, int out_size, void* d_ws, size_t ws_size,
                              hipStream_t stream) {
    const float* x   = (const float*)d_in[0];
    const float* w_q = (const float*)d_in[1];
    const float* w_k = (const float*)d_in[2];
    const float* w_v = (const float*)d_in[3];
    const float* w_o = (const float*)d_in[4];
    const float* kc  = (const float*)d_in[5];
    const float* vc  = (const float*)d_in[6];
    const int* dlens = (const int*)d_in[8];
    const int* didx  = (const int*)d_in[9];
    const int* plens = (const int*)d_in[11];

    const int NT = in_sizes[0] / HID_;       // 2056
    const int ND = in_sizes[8];              // 8
    const int NP = in_sizes[11];             // 4
    const int TQ = NT - ND;                  // 2048 prefill tokens

    float* out = (float*)d_out;
    const size_t tok = (size_t)NT * HID_;
    float* Qb = (float*)d_ws;
    float* Kb = Qb + tok;
    float* Vb = Kb + tok;
    float* Ob = Vb + tok;

    const float scale = 1.0f / sqrtf((float)DH_);

    dim3 gblk(256);
    dim3 ggrd(HID_ / GEMM_BN, (NT + GEMM_BM - 1) / GEMM_BM);

    // Q/K/V projections
    gemm_xwt_kernel<<<ggrd, gblk, 0, stream>>>(x, w_q, Qb, NT, HID_, HID_);
    gemm_xwt_kernel<<<ggrd, gblk, 0, stream>>>(x, w_k, Kb, NT, HID_, HID_);
    gemm_xwt_kernel<<<ggrd, gblk, 0, stream>>>(x, w_v, Vb, NT, HID_, HID_);

    // decode attention -> Ob rows [0, ND)
    decode_attn_kernel<<<dim3(ND * H_), dim3(128), 0, stream>>>(
        Qb, Kb, Vb, kc, vc, dlens, didx, Ob, scale);

    // prefill attention -> Ob rows [ND, NT)
    prefill_attn_kernel<<<dim3(TQ / 16, H_), dim3(32), 0, stream>>>(
        Qb, Kb, Vb, plens, NP, ND, Ob, scale);

    // output projection
    gemm_xwt_kernel<<<ggrd, gblk, 0, stream>>>(Ob, w_o, out, NT, HID_, HID_);
}